// FewShotLearningModule_89206470738094
// MI455X (gfx1250) — compile-verified
//
#include <hip/hip_runtime.h>
#include <hip/hip_bf16.h>
#include <math.h>

typedef float v2f __attribute__((ext_vector_type(2)));
typedef float v8f __attribute__((ext_vector_type(8)));

#define HDIM 1024
#define BATCH 64
#define SEQ 512

// workspace layout (float offsets)
#define WS_PARTIAL 0                       // 64*8*1024 = 524288
#define WS_POOLED  (524288)                // 65536
#define WS_TMETA   (524288 + 65536)        // 65536
#define WS_SIMS    (524288 + 131072)       // 64*100 = 6400
#define WS_PRED    (524288 + 131072 + 6400)// 64*3

// ---------------------------------------------------------------------------
// Stage 1a: partial mean over sequence. grid = 64*8 blocks, 256 threads.
// block handles (b, s-chunk of 64). float4 coalesced: 256 thr * 16B = 4KB row.
// ---------------------------------------------------------------------------
__global__ __launch_bounds__(256)
void fsl_pool_partial(const float* __restrict__ hs, float* __restrict__ partial) {
    const int b  = blockIdx.x >> 3;
    const int sc = blockIdx.x & 7;
    const size_t base = ((size_t)b * SEQ + (size_t)sc * 64) * HDIM;
    const int off = threadIdx.x * 4;
    float4 acc = make_float4(0.f, 0.f, 0.f, 0.f);
    for (int s = 0; s < 64; ++s) {
        float4 v = *(const float4*)(hs + base + (size_t)s * HDIM + off);
        acc.x += v.x; acc.y += v.y; acc.z += v.z; acc.w += v.w;
    }
    *(float4*)(partial + (size_t)blockIdx.x * HDIM + off) = acc;
}

// ---------------------------------------------------------------------------
// Stage 1b: reduce 8 partials -> pooled[b][h] (scaled by 1/512). 64 blocks.
// ---------------------------------------------------------------------------
__global__ __launch_bounds__(256)
void fsl_pool_reduce(const float* __restrict__ partial, float* __restrict__ pooled) {
    const int b = blockIdx.x;
    const int off = threadIdx.x * 4;
    float4 acc = make_float4(0.f, 0.f, 0.f, 0.f);
    for (int sc = 0; sc < 8; ++sc) {
        float4 v = *(const float4*)(partial + (size_t)(b * 8 + sc) * HDIM + off);
        acc.x += v.x; acc.y += v.y; acc.z += v.z; acc.w += v.w;
    }
    const float inv = 1.0f / (float)SEQ;
    acc.x *= inv; acc.y *= inv; acc.z *= inv; acc.w *= inv;
    *(float4*)(pooled + (size_t)b * HDIM + off) = acc;
}

// ---------------------------------------------------------------------------
// f32 WMMA 16x16 tile, K accumulated in steps of 4 (V_WMMA_F32_16X16X4_F32).
// A layout: lane m=lane&15 supplies A[m][k + 2*half + {0,1}] in 2 VGPRs.
// B layout: lane n=lane&15 supplies B[k + 2*half + {0,1}][n] = Wrow_n[k+...].
// C/D: VGPR r <-> row (r + 8*half), col = lane&15.
// ---------------------------------------------------------------------------
__device__ __forceinline__ v8f fsl_wmma_tile(const float* __restrict__ pA,
                                             const float* __restrict__ pB,
                                             int hoff, float bias) {
    v8f c;
#pragma unroll
    for (int r = 0; r < 8; ++r) c[r] = bias;
#pragma unroll 8
    for (int k = 0; k < HDIM; k += 4) {
        v2f a = *(const v2f*)(pA + k + hoff);
        v2f b = *(const v2f*)(pB + k + hoff);
        c = __builtin_amdgcn_wmma_f32_16x16x4_f32(false, a, false, b,
                                                  (short)0, c, false, false);
    }
    return c;
}

// ---------------------------------------------------------------------------
// Stage 2: all GEMMs that depend only on pooled. One wave per 16x16 tile.
// tiles: [0,32)   sim   = pooled @ Wsim^T  + bsim   -> out_sim  [64,128]
//        [32,64)  div   = pooled @ Wdiv^T  + bdiv   -> out_div  [64,128]
//        [64,320) tmeta = pooled @ Wmeta^T + bmeta  -> ws       [64,1024]
//        [320,348) sims = pooled @ supF'^T          -> ws       [64,100]
//                  (supF' rows 0..63 = pooled, rows 64..99 = support_features)
// ---------------------------------------------------------------------------
__global__ __launch_bounds__(32)
void fsl_gemm_stage1(const float* __restrict__ pooled,
                     const float* __restrict__ supF,
                     const float* __restrict__ Wsim, const float* __restrict__ bsim,
                     const float* __restrict__ Wdiv, const float* __restrict__ bdiv,
                     const float* __restrict__ Wmeta, const float* __restrict__ bmeta,
                     float* __restrict__ out_sim,
                     float* __restrict__ out_div,
                     float* __restrict__ tmeta,
                     float* __restrict__ sims) {
    const int lane = threadIdx.x;
    const int ml   = lane & 15;
    const int half = lane >> 4;
    const int hoff = half << 1;

    int t = blockIdx.x;
    int tm, tn, ldo, ntotal;
    const float* Brow;
    float bias;
    float* outp;

    if (t < 32) {
        tm = t >> 3; tn = t & 7;
        const int n = tn * 16 + ml;
        Brow = Wsim + (size_t)n * HDIM; bias = bsim[n];
        outp = out_sim; ldo = 128; ntotal = 128;
    } else if (t < 64) {
        t -= 32; tm = t >> 3; tn = t & 7;
        const int n = tn * 16 + ml;
        Brow = Wdiv + (size_t)n * HDIM; bias = bdiv[n];
        outp = out_div; ldo = 128; ntotal = 128;
    } else if (t < 320) {
        t -= 64; tm = t >> 6; tn = t & 63;
        const int n = tn * 16 + ml;
        Brow = Wmeta + (size_t)n * HDIM; bias = bmeta[n];
        outp = tmeta; ldo = HDIM; ntotal = HDIM;
    } else {
        t -= 320; tm = t / 7; tn = t % 7;
        const int n = tn * 16 + ml;
        if (n < 64)        Brow = pooled + (size_t)n * HDIM;   // updated support set
        else if (n < 100)  Brow = supF   + (size_t)n * HDIM;
        else               Brow = supF   + (size_t)64 * HDIM;  // dummy valid row
        bias = 0.0f;
        outp = sims; ldo = 100; ntotal = 100;
    }

    const float* pA = pooled + (size_t)(tm * 16 + ml) * HDIM;
    v8f c = fsl_wmma_tile(pA, Brow, hoff, bias);

    const int ng = tn * 16 + ml;
    if (ng < ntotal) {
        const int mbase = tm * 16 + half * 8;
#pragma unroll
        for (int r = 0; r < 8; ++r)
            outp[(size_t)(mbase + r) * ldo + ng] = c[r];
    }
}

// ---------------------------------------------------------------------------
// Stage 3: con = tmeta @ Wcon^T + bcon -> out_con [64,256]. 64 tiles.
// ---------------------------------------------------------------------------
__global__ __launch_bounds__(32)
void fsl_gemm_stage2(const float* __restrict__ tmeta,
                     const float* __restrict__ Wcon, const float* __restrict__ bcon,
                     float* __restrict__ out_con) {
    const int lane = threadIdx.x;
    const int ml   = lane & 15;
    const int half = lane >> 4;
    const int hoff = half << 1;

    const int tm = blockIdx.x >> 4;   // 0..3
    const int tn = blockIdx.x & 15;   // 0..15
    const int n  = tn * 16 + ml;

    const float* pA = tmeta + (size_t)(tm * 16 + ml) * HDIM;
    const float* pB = Wcon + (size_t)n * HDIM;
    v8f c = fsl_wmma_tile(pA, pB, hoff, bcon[n]);

    const int mbase = tm * 16 + half * 8;
#pragma unroll
    for (int r = 0; r < 8; ++r)
        out_con[(size_t)(mbase + r) * 256 + n] = c[r];
}

// ---------------------------------------------------------------------------
// Stage 4: per-row top-5 over sims[b][0..99], softmax, one-hot weighted sum.
// Labels of support set: rows 0..63 were overwritten with `labels`.
// One thread per batch row (64 rows of 100 values — trivial).
// ---------------------------------------------------------------------------
__global__ __launch_bounds__(64)
void fsl_topk_pred(const float* __restrict__ sims,
                   const int* __restrict__ labels,
                   const int* __restrict__ slabels,
                   float* __restrict__ pred) {
    const int b = threadIdx.x;
    if (b >= BATCH) return;

    float v[5]; int ix[5];
#pragma unroll
    for (int i = 0; i < 5; ++i) { v[i] = -INFINITY; ix[i] = 0; }

    for (int j = 0; j < 100; ++j) {
        const float s = sims[b * 100 + j];
        if (s > v[4]) {                       // strict: ties keep earliest index
            v[4] = s; ix[4] = j;
            for (int p = 4; p > 0 && v[p] > v[p - 1]; --p) {
                float tv = v[p]; v[p] = v[p - 1]; v[p - 1] = tv;
                int ti = ix[p]; ix[p] = ix[p - 1]; ix[p - 1] = ti;
            }
        }
    }

    const float mx = v[0];
    float e[5], sum = 0.f;
#pragma unroll
    for (int i = 0; i < 5; ++i) { e[i] = expf(v[i] - mx); sum += e[i]; }
    const float inv = 1.0f / sum;

    float p0 = 0.f, p1 = 0.f, p2 = 0.f;
#pragma unroll
    for (int i = 0; i < 5; ++i) {
        const int idx = ix[i];
        const int lab = (idx < BATCH) ? labels[idx] : slabels[idx];
        const float w = e[i] * inv;
        if (lab == 0) p0 += w;
        else if (lab == 1) p1 += w;
        else if (lab == 2) p2 += w;           // out-of-range label contributes 0
    }
    pred[b * 3 + 0] = p0;
    pred[b * 3 + 1] = p1;
    pred[b * 3 + 2] = p2;
}

// ---------------------------------------------------------------------------
// Stage 5: broadcast pred[b][c] -> few_shot_predictions[b][s][c]  (64*512*3)
// ---------------------------------------------------------------------------
__global__ __launch_bounds__(256)
void fsl_broadcast(const float* __restrict__ pred, float* __restrict__ out) {
    const int i = blockIdx.x * 256 + threadIdx.x;
    if (i < BATCH * SEQ * 3) {
        const int b = i / (SEQ * 3);
        const int c = i % 3;
        out[i] = pred[b * 3 + c];
    }
}

// ---------------------------------------------------------------------------
extern "C" void kernel_launch(void* const* d_in, const int* in_sizes, int n_in,
                              void* d_out, int out_size, void* d_ws, size_t ws_size,
                              hipStream_t stream) {
    const float* hs     = (const float*)d_in[0];   // [64,512,1024]
    const int*   labels = (const int*)  d_in[1];   // [64]
    const float* supF   = (const float*)d_in[2];   // [100,1024]
    const int*   supL   = (const int*)  d_in[3];   // [100]
    const float* Wsim   = (const float*)d_in[4];
    const float* bsim   = (const float*)d_in[5];
    const float* Wdiv   = (const float*)d_in[6];
    const float* bdiv   = (const float*)d_in[7];
    const float* Wmeta  = (const float*)d_in[8];
    const float* bmeta  = (const float*)d_in[9];
    const float* Wcon   = (const float*)d_in[10];
    const float* bcon   = (const float*)d_in[11];

    float* ws      = (float*)d_ws;
    float* partial = ws + WS_PARTIAL;
    float* pooled  = ws + WS_POOLED;
    float* tmeta   = ws + WS_TMETA;
    float* sims    = ws + WS_SIMS;
    float* pred    = ws + WS_PRED;

    float* out     = (float*)d_out;
    float* out_fsp = out;               // [64,512,3]  = 98304
    float* out_sim = out + 98304;       // [64,128]
    float* out_div = out + 106496;      // [64,128]
    float* out_con = out + 114688;      // [64,256]

    fsl_pool_partial<<<BATCH * 8, 256, 0, stream>>>(hs, partial);
    fsl_pool_reduce <<<BATCH,     256, 0, stream>>>(partial, pooled);
    fsl_gemm_stage1 <<<348, 32, 0, stream>>>(pooled, supF,
                                             Wsim, bsim, Wdiv, bdiv, Wmeta, bmeta,
                                             out_sim, out_div, tmeta, sims);
    fsl_gemm_stage2 <<<64, 32, 0, stream>>>(tmeta, Wcon, bcon, out_con);
    fsl_topk_pred   <<<1, 64, 0, stream>>>(sims, labels, supL, pred);
    fsl_broadcast   <<<(BATCH * SEQ * 3 + 255) / 256, 256, 0, stream>>>(pred, out_fsp);
}